// RPNRefine_67078799229102
// MI455X (gfx1250) — compile-verified
//
#include <hip/hip_runtime.h>
#include <stdint.h>

// ---------------------------------------------------------------------------
// RPN proposal generation (top-k + box refine + greedy NMS) for MI455X.
// No dense GEMM exists in this workload (IoU = min/max algebra), so WMMA is
// inapplicable. CDNA5-specific paths used instead:
//   * TDM: tensor_load_to_lds (D# descriptor DMA) + s_wait_tensorcnt for the
//     bulk 96KB box tile staging in NMS
//   * global_load_async_to_lds_b64 / global_store_async_from_lds_b64
//     (ASYNCcnt engine) for key staging in the bitonic sort
//   * 320KB-per-WGP LDS: entire 6000-box NMS working set resident on-chip
//   * wave32 __ballot for the serial NMS "next survivor" scan
//   * global_prefetch_b8 on the streaming histogram pass
// ---------------------------------------------------------------------------

#define BATCH   8
#define NA      261888
#define PRE     6000
#define PADN    6016          // padded candidate rows (16B each)
#define POST    1000
#define SORTN   8192
#define NB      4096
#define IMGD    1024.0f
#define IOU_THR 0.7f
#define NQW     (PRE * 2)     // box tile size in 8-byte units (12000)

typedef uint32_t v4u __attribute__((ext_vector_type(4)));
typedef uint32_t v8u __attribute__((ext_vector_type(8)));

// ---- scratch layout (bytes) ----
#define OFF_HIST   0u                         // BATCH*NB*4      = 131072
#define OFF_HIST2  131072u                    // BATCH*NB*4      = 131072
#define OFF_META   262144u                    // BATCH*16*4      = 512
#define OFF_KEYS   262656u                    // BATCH*SORTN*8   = 524288
#define OFF_BOXES  786944u                    // BATCH*PADN*16   = 770048
#define ZERO_U32   196736                     // u32 count covering hist..keys

// ---------------------------------------------------------------------------
__global__ void k_zero(uint32_t* __restrict__ p, int n) {
  int i = blockIdx.x * blockDim.x + threadIdx.x;
  if (i < n) p[i] = 0u;
}

// Level-1 histogram over score bits [30:19] (softmax scores >= 0, so IEEE
// bit order == numeric order).
__global__ void k_hist1(const float* __restrict__ cls, uint32_t* __restrict__ hist) {
  __shared__ uint32_t h[NB];
  const int b = blockIdx.y;
  for (int t = threadIdx.x; t < NB; t += blockDim.x) h[t] = 0u;
  __syncthreads();
  const float* s = cls + (size_t)b * NA * 2;
  const int span = (NA + gridDim.x - 1) / gridDim.x;
  const int lo = blockIdx.x * span;
  const int hi = min(lo + span, NA);
  for (int i = lo + threadIdx.x; i < hi; i += blockDim.x) {
    __builtin_prefetch(&s[2 * i + 2048], 0, 0);       // -> global_prefetch_b8
    uint32_t bits = __float_as_uint(s[2 * i + 1]);
    atomicAdd(&h[bits >> 19], 1u);
  }
  __syncthreads();
  uint32_t* g = hist + (size_t)b * NB;
  for (int t = threadIdx.x; t < NB; t += blockDim.x)
    if (h[t]) atomicAdd(&g[t], h[t]);
}

// Find cut bucket: cumulative-from-top first reaches >= PRE.
__global__ void k_cut1(const uint32_t* __restrict__ hist, uint32_t* __restrict__ meta) {
  if (threadIdx.x != 0) return;
  const int b = blockIdx.x;
  const uint32_t* h = hist + (size_t)b * NB;
  uint32_t acc = 0, above = 0, cut = 0;
  for (int bk = NB - 1; bk >= 0; --bk) {
    uint32_t prev = acc;
    acc += h[bk];
    if (acc >= PRE) { cut = (uint32_t)bk; above = prev; break; }
  }
  uint32_t* m = meta + b * 16;
  m[0] = cut;     // level-1 cut bucket
  m[1] = above;   // count strictly above cut bucket
}

// Level-2 histogram (bits [18:7]) restricted to the level-1 cut bucket.
__global__ void k_hist2(const float* __restrict__ cls,
                        const uint32_t* __restrict__ meta,
                        uint32_t* __restrict__ hist2) {
  __shared__ uint32_t h[NB];
  const int b = blockIdx.y;
  const uint32_t cut1 = meta[b * 16 + 0];
  for (int t = threadIdx.x; t < NB; t += blockDim.x) h[t] = 0u;
  __syncthreads();
  const float* s = cls + (size_t)b * NA * 2;
  const int span = (NA + gridDim.x - 1) / gridDim.x;
  const int lo = blockIdx.x * span;
  const int hi = min(lo + span, NA);
  for (int i = lo + threadIdx.x; i < hi; i += blockDim.x) {
    uint32_t bits = __float_as_uint(s[2 * i + 1]);
    if ((bits >> 19) == cut1) atomicAdd(&h[(bits >> 7) & 0xFFFu], 1u);
  }
  __syncthreads();
  uint32_t* g = hist2 + (size_t)b * NB;
  for (int t = threadIdx.x; t < NB; t += blockDim.x)
    if (h[t]) atomicAdd(&g[t], h[t]);
}

__global__ void k_cut2(const uint32_t* __restrict__ hist2, uint32_t* __restrict__ meta) {
  if (threadIdx.x != 0) return;
  const int b = blockIdx.x;
  uint32_t* m = meta + b * 16;
  const uint32_t* h = hist2 + (size_t)b * NB;
  const uint32_t need = PRE - m[1];          // >=1 by construction
  uint32_t acc = 0, cut2 = 0;
  for (int bk = NB - 1; bk >= 0; --bk) {
    acc += h[bk];
    if (acc >= need) { cut2 = (uint32_t)bk; break; }
  }
  m[2] = cut2;
}

// Compact candidates: key = score_bits<<32 | ~index. Descending sort then
// yields exact (score desc, index asc) order, matching lax.top_k ties.
__global__ void k_compact(const float* __restrict__ cls,
                          uint32_t* __restrict__ meta,
                          uint64_t* __restrict__ keys) {
  const int b = blockIdx.y;
  uint32_t* m = meta + b * 16;
  const uint32_t cut1 = m[0], cut2 = m[2];
  const float* s = cls + (size_t)b * NA * 2;
  uint64_t* kb = keys + (size_t)b * SORTN;
  const int span = (NA + gridDim.x - 1) / gridDim.x;
  const int lo = blockIdx.x * span;
  const int hi = min(lo + span, NA);
  for (int i = lo + threadIdx.x; i < hi; i += blockDim.x) {
    uint32_t bits = __float_as_uint(s[2 * i + 1]);
    uint32_t bk = bits >> 19;
    bool cand = (bk > cut1) || (bk == cut1 && ((bits >> 7) & 0xFFFu) >= cut2);
    if (cand) {
      uint32_t pos = atomicAdd(&m[3], 1u);
      if (pos < SORTN)
        kb[pos] = ((uint64_t)bits << 32) | (uint64_t)(0xFFFFFFFFu - (uint32_t)i);
    }
  }
}

// Bitonic sort 8192 u64 keys, descending, in 64KB dynamic LDS; one block/image.
// Keys staged in AND out with the CDNA5 async global<->LDS engine (no VGPR
// round-trip for the bulk data).
__global__ __launch_bounds__(1024) void k_sort(uint64_t* __restrict__ keys) {
  extern __shared__ uint64_t sk[];
  const int b = blockIdx.x;
  const uint64_t gbase = (uint64_t)(void*)(keys + (size_t)b * SORTN);
  const uint32_t lbase = (uint32_t)(size_t)(void*)&sk[0];

  for (int t = threadIdx.x; t < SORTN; t += blockDim.x) {
    uint32_t loff = lbase + (uint32_t)t * 8u;
    uint64_t gadr = gbase + (uint64_t)t * 8u;
    asm volatile("global_load_async_to_lds_b64 %0, %1, off"
                 :: "v"(loff), "v"(gadr) : "memory");
  }
  asm volatile("s_wait_asynccnt 0x0" ::: "memory");
  __syncthreads();

  for (int k = 2; k <= SORTN; k <<= 1) {
    for (int j = k >> 1; j > 0; j >>= 1) {
      for (int t = threadIdx.x; t < SORTN; t += blockDim.x) {
        int ixj = t ^ j;
        if (ixj > t) {
          uint64_t a = sk[t], c = sk[ixj];
          bool sw = ((t & k) == 0) ? (a < c) : (a > c);   // descending
          if (sw) { sk[t] = c; sk[ixj] = a; }
        }
      }
      __syncthreads();   // also forces s_wait_dscnt: LDS committed for async rd
    }
  }

  for (int t = threadIdx.x; t < PRE; t += blockDim.x) {
    uint32_t loff = lbase + (uint32_t)t * 8u;
    uint64_t gadr = gbase + (uint64_t)t * 8u;
    asm volatile("global_store_async_from_lds_b64 %0, %1, off"
                 :: "v"(gadr), "v"(loff) : "memory");
  }
  asm volatile("s_wait_asynccnt 0x0" ::: "memory");
}

// Gather anchors/deltas for the sorted top-PRE, apply deltas, clip, store AoS.
__global__ void k_refine(const uint64_t* __restrict__ keys,
                         const float* __restrict__ bbox,
                         const float* __restrict__ anchors,
                         float4* __restrict__ boxes) {
  const int b = blockIdx.y;
  const int k = blockIdx.x * blockDim.x + threadIdx.x;
  if (k >= PRE) return;
  uint64_t key = keys[(size_t)b * SORTN + k];
  uint32_t idx = 0xFFFFFFFFu - (uint32_t)(key & 0xFFFFFFFFull);
  if (idx >= NA) idx = 0;
  const float* a = anchors + (size_t)idx * 4;
  const float* d = bbox + ((size_t)b * NA + (size_t)idx) * 4;
  float y1 = a[0], x1 = a[1], y2 = a[2], x2 = a[3];
  float h = y2 - y1, w = x2 - x1;
  float cy = y1 + 0.5f * h + d[0] * 0.1f * h;
  float cx = x1 + 0.5f * w + d[1] * 0.1f * w;
  h *= __expf(d[2] * 0.2f);
  w *= __expf(d[3] * 0.2f);
  float4 out;
  out.x = fminf(fmaxf(cy - 0.5f * h, 0.f), IMGD);
  out.y = fminf(fmaxf(cx - 0.5f * w, 0.f), IMGD);
  out.z = fminf(fmaxf(cy + 0.5f * h, 0.f), IMGD);
  out.w = fminf(fmaxf(cx + 0.5f * w, 0.f), IMGD);
  boxes[(size_t)b * PADN + k] = out;
}

// Greedy NMS, one workgroup per image, all boxes resident in LDS.
// The 96KB box tile is staged global->LDS with ONE Tensor Data Mover
// descriptor issued by wave 0 (TENSORcnt), freeing all 32 waves instantly.
#define NMS_SMEM (PADN * 16 + PADN + POST * 4 + 16)
__global__ __launch_bounds__(1024) void k_nms(const float4* __restrict__ boxes,
                                              float* __restrict__ out) {
  extern __shared__ uint8_t smem[];
  float4*  sbox   = (float4*)smem;                         // PADN*16
  uint8_t* ssup   = smem + PADN * 16;                      // PADN
  int*     s_kept = (int*)(smem + PADN * 16 + PADN);       // POST*4
  int*     s_ctl  = (int*)(smem + PADN * 16 + PADN + POST * 4); // [0]=next [1]=cnt

  const int b    = blockIdx.x;
  const int tid  = threadIdx.x;
  const int lane = tid & 31;                 // wave32
  const uint64_t gbase = (uint64_t)(const void*)(boxes + (size_t)b * PADN);
  // low 32 bits of the generic shared address = LDS byte offset for this WG
  const uint32_t lbase = (uint32_t)(size_t)(void*)&sbox[0];

  // ---- TDM: one D#-descriptor DMA for the whole 6000x16B tile ----
  if ((tid >> 5) == 0) {                     // wave 0 only (EXEC ignored by TDM)
    // D# group 0: count=1 | lds_addr | global_addr | type=2
    v4u g0 = { 1u,
               lbase,
               (uint32_t)gbase,
               (((uint32_t)(gbase >> 32)) & 0x01FFFFFFu) | 0x80000000u };
    // D# group 1: mask=0, data_size=3 (8B); 1-D tensor/tile of NQW elements
    v8u g1 = { 3u << 16,                         // [17:16] data_size=8B
               ((uint32_t)NQW & 0xFFFFu) << 16,  // tensor_dim0[15:0] @[63:48]
               (((uint32_t)NQW >> 16) & 0xFFFFu) // tensor_dim0[31:16] @[79:64]
                 | (1u << 16),                   // tensor_dim1=1     @[111:80]
               ((uint32_t)NQW & 0xFFFFu) << 16,  // tile_dim0         @[127:112]
               0u,                               // tile_dim1/2 unused
               (uint32_t)NQW,                    // tensor_dim0_stride[31:0]
               0u, 0u };
    asm volatile("tensor_load_to_lds %0, %1" :: "s"(g0), "s"(g1) : "memory");
    __builtin_amdgcn_s_wait_tensorcnt(0);        // -> s_wait_tensorcnt 0x0
  }
  for (int j = tid; j < PADN; j += blockDim.x) ssup[j] = (j < PRE) ? 0 : 1;
  if (tid == 0) s_ctl[1] = 0;
  __syncthreads();

  int cur = 0;
  while (true) {
    if (tid == 0) s_ctl[0] = 0x7FFFFFFF;
    __syncthreads();
    {
      int j = cur + tid;
      bool p = (j < PRE) && (ssup[j] == 0);
      unsigned long long m = __ballot(p);        // wave32: mask in low 32 bits
      if (lane == 0 && m != 0ull) {
        int first = __ffsll((unsigned long long)m) - 1;
        atomicMin(&s_ctl[0], cur + (tid & ~31) + first);
      }
    }
    __syncthreads();
    int i = s_ctl[0];
    if (i == 0x7FFFFFFF) {             // whole window suppressed -> advance
      cur += blockDim.x;
      if (cur >= PRE) break;
      continue;
    }
    int kc = s_ctl[1];
    __syncthreads();
    if (tid == 0) { s_kept[kc] = i; s_ctl[1] = kc + 1; ssup[i] = 1; }
    __syncthreads();
    if (kc + 1 >= POST) break;

    float4 bi = sbox[i];
    float areaI = (bi.z - bi.x) * (bi.w - bi.y);
    for (int j = i + 1 + tid; j < PRE; j += blockDim.x) {
      if (ssup[j]) continue;
      float4 bj = sbox[j];
      float ih = fmaxf(fminf(bi.z, bj.z) - fmaxf(bi.x, bj.x), 0.f);
      float iw = fmaxf(fminf(bi.w, bj.w) - fmaxf(bi.y, bj.y), 0.f);
      float inter = ih * iw;
      float areaJ = (bj.z - bj.x) * (bj.w - bj.y);
      float iou = inter / (areaI + areaJ - inter + 1e-8f);
      if (iou > IOU_THR) ssup[j] = 1;
    }
    cur = i + 1;
  }
  __syncthreads();

  const int cnt = s_ctl[1];
  float4* op = (float4*)(out + (size_t)b * POST * 4);
  for (int k = tid; k < POST; k += blockDim.x) {
    float4 v = make_float4(0.f, 0.f, 0.f, 0.f);
    if (k < cnt) v = sbox[(unsigned)s_kept[k]];
    op[k] = v;
  }
}

// ---------------------------------------------------------------------------
extern "C" void kernel_launch(void* const* d_in, const int* in_sizes, int n_in,
                              void* d_out, int out_size, void* d_ws, size_t ws_size,
                              hipStream_t stream) {
  (void)in_sizes; (void)n_in; (void)out_size; (void)ws_size;
  const float* cls     = (const float*)d_in[0];   // B x NA x 2
  const float* bbox    = (const float*)d_in[1];   // B x NA x 4
  const float* anchors = (const float*)d_in[2];   // NA x 4

  uint8_t*  ws    = (uint8_t*)d_ws;               // needs ~1.56 MB
  uint32_t* hist  = (uint32_t*)(ws + OFF_HIST);
  uint32_t* hist2 = (uint32_t*)(ws + OFF_HIST2);
  uint32_t* meta  = (uint32_t*)(ws + OFF_META);
  uint64_t* keys  = (uint64_t*)(ws + OFF_KEYS);
  float4*   boxes = (float4*)(ws + OFF_BOXES);

  k_zero<<<(ZERO_U32 + 255) / 256, 256, 0, stream>>>((uint32_t*)ws, ZERO_U32);
  k_hist1<<<dim3(64, BATCH), 256, 0, stream>>>(cls, hist);
  k_cut1<<<BATCH, 32, 0, stream>>>(hist, meta);
  k_hist2<<<dim3(64, BATCH), 256, 0, stream>>>(cls, meta, hist2);
  k_cut2<<<BATCH, 32, 0, stream>>>(hist2, meta);
  k_compact<<<dim3(64, BATCH), 256, 0, stream>>>(cls, meta, keys);
  k_sort<<<BATCH, 1024, SORTN * sizeof(uint64_t), stream>>>(keys);
  k_refine<<<dim3((PRE + 255) / 256, BATCH), 256, 0, stream>>>(keys, bbox, anchors, boxes);
  k_nms<<<BATCH, 1024, NMS_SMEM, stream>>>(boxes, (float*)d_out);
}